// GeoConvNet3DMeshSeg_8323646619910
// MI455X (gfx1250) — compile-verified
//
#include <hip/hip_runtime.h>
#include <hip/hip_bf16.h>

typedef float v2f __attribute__((ext_vector_type(2)));
typedef float v8f __attribute__((ext_vector_type(8)));

#define BN_EPS 1e-5f

// ---------------------------------------------------------------------------
// Fused mesh_conv GEMM: out(E x COUT) = feat(E x 5C) @ W(5C x COUT)
// feat row e = [x[e], min(n0,n1), max(n0,n1), min(n2,n3), max(n2,n3)]
// Block: 16 rows x 64 cols, 4 waves (wave32), one 16x16 f32 WMMA tile/wave.
// A chunk (16 x KC) staged in LDS with the gather+min/max fused in.
// Compile-time C and COUT -> immediate-offset W loads and output stores.
// ---------------------------------------------------------------------------
template <int C, int COUT>
__global__ __launch_bounds__(128) void meshconv_gemm(
    const float* __restrict__ x, const int* __restrict__ nb, int E,
    const float* __restrict__ W, float* __restrict__ out)
{
    constexpr int K  = 5 * C;
    constexpr int KC = (K < 64) ? 32 : 64;   // K-chunk per LDS stage
    constexpr int AS = KC + 2;               // padded row stride (8B aligned)
    __shared__ float Atile[16 * AS];
    __shared__ int   nbs[16 * 4];            // clamped neighbor ids for this tile

    const int row0 = blockIdx.x * 16;
    const int tid  = threadIdx.x;
    const int wave = tid >> 5;
    const int lane = tid & 31;
    const int ln   = lane & 15;
    const int hi   = lane >> 4;
    const int ncol = blockIdx.y * 64 + wave * 16 + ln;   // this lane's out column

    // stage clamped neighbor indices once per block
    if (tid < 64) {
        int r = tid >> 2;
        int m = row0 + r;
        int t = (m < E) ? nb[m * 4 + (tid & 3)] : 0;
        t = t < 0 ? 0 : (t > E - 1 ? E - 1 : t);
        nbs[tid] = t;
    }
    const bool fullTile = (row0 + 16 <= E);

    v8f acc = {0.f, 0.f, 0.f, 0.f, 0.f, 0.f, 0.f, 0.f};

    for (int k0 = 0; k0 < K; k0 += KC) {
        __syncthreads();
        // ---- stage A chunk: 16 rows x KC k-columns, gather fused ----
        #pragma unroll
        for (int i = 0; i < (16 * KC) / 128; ++i) {
            int idx = tid + i * 128;
            int r   = idx / KC;
            int kc  = idx % KC;
            int kk  = k0 + kc;
            int m   = row0 + r;
            float v = 0.0f;
            bool ok = (fullTile || m < E) && (K % KC == 0 || kk < K);
            if (ok) {
                int part = kk / C;           // 0..4 (constant divisor)
                int c    = kk - part * C;
                if (part == 0) {
                    v = x[(size_t)m * C + c];
                } else {
                    int j0 = (part <= 2) ? 0 : 2;
                    int n0 = nbs[r * 4 + j0];
                    int n1 = nbs[r * 4 + j0 + 1];
                    float va = x[(size_t)n0 * C + c];
                    float vb = x[(size_t)n1 * C + c];
                    v = (part & 1) ? fminf(va, vb) : fmaxf(va, vb);
                }
            }
            Atile[r * AS + kc] = v;
        }
        __syncthreads();

        if constexpr (K % KC == 0) {
            // unguarded fast path: full chunk, immediate-offset B loads
            const float* wp0 = W + (size_t)(k0 + (hi << 1)) * COUT + ncol;
            #pragma unroll
            for (int ks = 0; ks < KC; ks += 4) {
                int kb = ks + (hi << 1);
                v2f a;
                a.x = Atile[ln * AS + kb];
                a.y = Atile[ln * AS + kb + 1];
                v2f b;
                b.x = wp0[(size_t)ks * COUT];
                b.y = wp0[(size_t)(ks + 1) * COUT];
#if __has_builtin(__builtin_amdgcn_wmma_f32_16x16x4_f32)
                acc = __builtin_amdgcn_wmma_f32_16x16x4_f32(
                    false, a, false, b, (short)0, acc, false, false);
#else
                #pragma unroll
                for (int j = 0; j < 8; ++j) {
                    int r = j + (hi << 3);
                    float t = acc[j];
                    #pragma unroll
                    for (int q = 0; q < 4; ++q)
                        t += Atile[r * AS + ks + q] * W[(size_t)(k0+ks+q) * COUT + ncol];
                    acc[j] = t;
                }
#endif
            }
        } else {
            // guarded tail path (only e1: K=25)
            const int ksteps = (K - k0) < KC ? (K - k0) : KC;
            for (int ks = 0; ks < ksteps; ks += 4) {
                int kb  = ks + (hi << 1);
                int kwa = k0 + kb;
                v2f a;
                a.x = Atile[ln * AS + kb];
                a.y = Atile[ln * AS + kb + 1];
                v2f b;
                b.x = (kwa     < K) ? W[(size_t)kwa       * COUT + ncol] : 0.0f;
                b.y = (kwa + 1 < K) ? W[(size_t)(kwa + 1) * COUT + ncol] : 0.0f;
#if __has_builtin(__builtin_amdgcn_wmma_f32_16x16x4_f32)
                acc = __builtin_amdgcn_wmma_f32_16x16x4_f32(
                    false, a, false, b, (short)0, acc, false, false);
#else
                #pragma unroll
                for (int j = 0; j < 8; ++j) {
                    int r = j + (hi << 3);
                    float t = acc[j];
                    #pragma unroll
                    for (int q = 0; q < 4; ++q) {
                        int kw = k0 + ks + q;
                        float wv = (kw < K) ? W[(size_t)kw * COUT + ncol] : 0.0f;
                        t += Atile[r * AS + ks + q] * wv;
                    }
                    acc[j] = t;
                }
#endif
            }
        }
    }

    // D layout: lanes 0-15 rows 0..7, lanes 16-31 rows 8..15; col = ln.
    float* op = out + (size_t)(row0 + (hi << 3)) * COUT + ncol;
    if (fullTile) {
        #pragma unroll
        for (int j = 0; j < 8; ++j) op[(size_t)j * COUT] = acc[j];
    } else {
        #pragma unroll
        for (int j = 0; j < 8; ++j) {
            if (row0 + (hi << 3) + j < E) op[(size_t)j * COUT] = acc[j];
        }
    }
}

// ---- deterministic per-column batch stats (sum, sumsq) ----
__global__ __launch_bounds__(256) void col_stats_kernel(
    const float* __restrict__ h, int E, int COUT, float* __restrict__ sums)
{
    __shared__ float ss[256];
    __shared__ float sq[256];
    int c = blockIdx.x;
    float a = 0.f, b = 0.f;
    for (int r = threadIdx.x; r < E; r += 256) {
        float v = h[(size_t)r * COUT + c];
        a += v; b += v * v;
    }
    ss[threadIdx.x] = a; sq[threadIdx.x] = b;
    __syncthreads();
    for (int o = 128; o > 0; o >>= 1) {
        if (threadIdx.x < o) {
            ss[threadIdx.x] += ss[threadIdx.x + o];
            sq[threadIdx.x] += sq[threadIdx.x + o];
        }
        __syncthreads();
    }
    if (threadIdx.x == 0) { sums[c] = ss[0]; sums[COUT + c] = sq[0]; }
}

__global__ void bn_relu_kernel(float* __restrict__ h, const float* __restrict__ sums,
                               const float* __restrict__ g, const float* __restrict__ b,
                               int E, int COUT)
{
    int i = blockIdx.x * blockDim.x + threadIdx.x;
    if (i >= E * COUT) return;
    int c = i % COUT;
    float Ef  = (float)E;
    float mu  = sums[c] / Ef;
    float var = sums[COUT + c] / Ef - mu * mu;
    float v = (h[i] - mu) * rsqrtf(var + BN_EPS) * g[c] + b[c];
    h[i] = fmaxf(v, 0.0f);
}

// ---- pooling: exact top-k by L2 norm (bit-monotonic keys, two-level radix) ----
__global__ void scores_kernel(const float* __restrict__ xin, int E, int C,
                              unsigned* __restrict__ keys)
{
    int e = blockIdx.x * blockDim.x + threadIdx.x;
    if (e >= E) return;
    const float* row = xin + (size_t)e * C;
    float s = 0.f;
    for (int c = 0; c < C; ++c) { float v = row[c]; s += v * v; }
    keys[e] = __float_as_uint(sqrtf(s));   // >=0 -> order-preserving bits
}

__global__ void hist_hi_kernel(const unsigned* __restrict__ keys, int E,
                               unsigned* __restrict__ hist)
{
    int e = blockIdx.x * blockDim.x + threadIdx.x;
    if (e >= E) return;
    atomicAdd(&hist[keys[e] >> 16], 1u);
}

__global__ void sel_hi_kernel(const unsigned* __restrict__ hist, int target,
                              unsigned* __restrict__ sel)
{
    unsigned cum = 0; int b = 65535;
    for (; b >= 0; --b) {
        unsigned c = hist[b];
        if (cum + c >= (unsigned)target) break;
        cum += c;
    }
    if (b < 0) b = 0;
    sel[0] = (unsigned)b;   // hi bin
    sel[1] = cum;           // count strictly greater than hi bin
}

__global__ void hist_lo_kernel(const unsigned* __restrict__ keys, int E,
                               unsigned* __restrict__ hist,
                               const unsigned* __restrict__ sel)
{
    int e = blockIdx.x * blockDim.x + threadIdx.x;
    if (e >= E) return;
    unsigned k = keys[e];
    if ((k >> 16) == sel[0]) atomicAdd(&hist[k & 0xFFFFu], 1u);
}

__global__ void sel_lo_kernel(const unsigned* __restrict__ hist, int target,
                              unsigned* __restrict__ sel)
{
    unsigned remaining = (unsigned)target - sel[1];
    unsigned cum = 0; int b = 65535;
    for (; b >= 0; --b) {
        unsigned c = hist[b];
        if (cum + c >= remaining) break;
        cum += c;
    }
    if (b < 0) b = 0;
    sel[2] = (unsigned)b;
    sel[3] = remaining - cum;              // #ties (==T) to keep, lowest index first
    sel[4] = (sel[0] << 16) | (unsigned)b; // full 32-bit threshold key T
}

__global__ void eq_flags_kernel(const unsigned* __restrict__ keys, int E,
                                const unsigned* __restrict__ sel,
                                unsigned* __restrict__ flags)
{
    int e = blockIdx.x * blockDim.x + threadIdx.x;
    if (e >= E) return;
    flags[e] = (keys[e] == sel[4]) ? 1u : 0u;
}

__global__ void keep_flags_kernel(const unsigned* __restrict__ keys, int E,
                                  const unsigned* __restrict__ sel,
                                  const unsigned* __restrict__ eqscan,
                                  unsigned* __restrict__ flags)
{
    int e = blockIdx.x * blockDim.x + threadIdx.x;
    if (e >= E) return;
    unsigned k = keys[e], T = sel[4];
    flags[e] = (k > T || (k == T && eqscan[e] < sel[3])) ? 1u : 0u;
}

// ---- generic exclusive scan (3-phase) ----
__global__ __launch_bounds__(256) void scan_block_kernel(
    const unsigned* __restrict__ in, unsigned* __restrict__ out,
    unsigned* __restrict__ bsums, int n)
{
    __shared__ unsigned tmp[256];
    int gid = blockIdx.x * 256 + threadIdx.x;
    unsigned v = (gid < n) ? in[gid] : 0u;
    tmp[threadIdx.x] = v;
    __syncthreads();
    for (int off = 1; off < 256; off <<= 1) {
        unsigned t = (threadIdx.x >= off) ? tmp[threadIdx.x - off] : 0u;
        __syncthreads();
        tmp[threadIdx.x] += t;
        __syncthreads();
    }
    if (gid < n) out[gid] = tmp[threadIdx.x] - v;   // exclusive
    if (threadIdx.x == 255) bsums[blockIdx.x] = tmp[255];
}

__global__ void scan_sums_kernel(unsigned* __restrict__ bsums, int nb)
{
    unsigned acc = 0;
    for (int i = 0; i < nb; ++i) { unsigned t = bsums[i]; bsums[i] = acc; acc += t; }
}

__global__ void scan_add_kernel(unsigned* __restrict__ out,
                                const unsigned* __restrict__ bsums, int n)
{
    int gid = blockIdx.x * 256 + threadIdx.x;
    if (gid < n) out[gid] += bsums[blockIdx.x];
}

__global__ void compact_kernel(const unsigned* __restrict__ flags,
                               const unsigned* __restrict__ scan, int E,
                               int* __restrict__ keepIdx, int* __restrict__ remap)
{
    int e = blockIdx.x * blockDim.x + threadIdx.x;
    if (e >= E) return;
    if (flags[e]) {
        int pos = (int)scan[e];
        keepIdx[pos] = e;
        remap[e] = pos;
    } else {
        remap[e] = -1;
    }
}

__global__ void nb_pool_kernel(const int* __restrict__ keepIdx,
                               const int* __restrict__ remap,
                               const int* __restrict__ nb_old, int E_old,
                               int target, int* __restrict__ nb_new)
{
    int i = blockIdx.x * blockDim.x + threadIdx.x;
    if (i >= target) return;
    int o = keepIdx[i];
    #pragma unroll
    for (int j = 0; j < 4; ++j) {
        int t = nb_old[o * 4 + j];
        t = t < 0 ? 0 : (t > E_old - 1 ? E_old - 1 : t);
        int r = remap[t];
        nb_new[i * 4 + j] = (r < 0) ? i : r;
    }
}

__global__ void pool_x_kernel(const float* __restrict__ xin,
                              const int* __restrict__ keepIdx,
                              int target, int C, float* __restrict__ xp)
{
    int idx = blockIdx.x * blockDim.x + threadIdx.x;
    if (idx >= target * C) return;
    int r = idx / C, c = idx - r * C;
    xp[idx] = xin[(size_t)keepIdx[r] * C + c];
}

// ---- unpool (nearest kept by index, ties -> lower) into concat buffer ----
__global__ void unpool_cat_kernel(const float* __restrict__ xc,
                                  const int* __restrict__ keep, int Kn,
                                  int Efine, int Cc, int Ctot,
                                  float* __restrict__ dst)
{
    int idx = blockIdx.x * blockDim.x + threadIdx.x;
    if (idx >= Efine * Cc) return;
    int e = idx / Cc, c = idx - e * Cc;
    int lo = 0, hi2 = Kn;
    while (lo < hi2) { int mid = (lo + hi2) >> 1; if (keep[mid] < e) lo = mid + 1; else hi2 = mid; }
    int left  = lo - 1; if (left < 0) left = 0; if (left > Kn - 1) left = Kn - 1;
    int right = lo;     if (right > Kn - 1) right = Kn - 1;
    int dl = e - keep[left];  if (dl < 0) dl = -dl;
    int dr = e - keep[right]; if (dr < 0) dr = -dr;
    int src = (dl <= dr) ? left : right;
    dst[(size_t)e * Ctot + c] = xc[(size_t)src * Cc + c];
}

__global__ void copy_offset_kernel(const float* __restrict__ src, int E, int Cs,
                                   int Ctot, int off, float* __restrict__ dst)
{
    int idx = blockIdx.x * blockDim.x + threadIdx.x;
    if (idx >= E * Cs) return;
    int e = idx / Cs, c = idx - e * Cs;
    dst[(size_t)e * Ctot + off + c] = src[idx];
}

__global__ void head_kernel(const float* __restrict__ d1, const float* __restrict__ Wh,
                            const float* __restrict__ bias, int E, float* __restrict__ out)
{
    int e = blockIdx.x * blockDim.x + threadIdx.x;
    if (e >= E) return;
    float a0 = bias[0], a1 = bias[1], a2 = bias[2], a3 = bias[3];
    const float* row = d1 + (size_t)e * 64;
    #pragma unroll
    for (int c = 0; c < 64; ++c) {
        float v = row[c];
        a0 += v * Wh[c * 4 + 0];
        a1 += v * Wh[c * 4 + 1];
        a2 += v * Wh[c * 4 + 2];
        a3 += v * Wh[c * 4 + 3];
    }
    out[(size_t)e * 4 + 0] = a0; out[(size_t)e * 4 + 1] = a1;
    out[(size_t)e * 4 + 2] = a2; out[(size_t)e * 4 + 3] = a3;
}

// ---------------------------------------------------------------------------
// Host orchestration
// ---------------------------------------------------------------------------
static void run_scan(const unsigned* in, unsigned* out, unsigned* bsums, int n,
                     hipStream_t s)
{
    int nb = (n + 255) / 256;
    scan_block_kernel<<<nb, 256, 0, s>>>(in, out, bsums, n);
    scan_sums_kernel<<<1, 1, 0, s>>>(bsums, nb);
    scan_add_kernel<<<nb, 256, 0, s>>>(out, bsums, n);
}

template <int C, int COUT>
static void run_conv(const float* x, const int* nb, int E,
                     const float* W, const float* g, const float* b,
                     float* out, float* sums, hipStream_t s)
{
    dim3 grid((unsigned)((E + 15) / 16), (unsigned)(COUT / 64));
    meshconv_gemm<C, COUT><<<grid, 128, 0, s>>>(x, nb, E, W, out);
    col_stats_kernel<<<COUT, 256, 0, s>>>(out, E, COUT, sums);
    int n = E * COUT;
    bn_relu_kernel<<<(n + 255) / 256, 256, 0, s>>>(out, sums, g, b, E, COUT);
}

static void run_pool(const float* xin, const int* nb_old, int E, int C, int target,
                     float* xp, int* nb_new, int* keepIdx, int* remap,
                     unsigned* keys, unsigned* hist, unsigned* sel,
                     unsigned* flags, unsigned* scanA, unsigned* scanB,
                     unsigned* bsums, hipStream_t s)
{
    int nt = (E + 255) / 256;
    scores_kernel<<<nt, 256, 0, s>>>(xin, E, C, keys);
    (void)hipMemsetAsync(hist, 0, 65536 * sizeof(unsigned), s);
    hist_hi_kernel<<<nt, 256, 0, s>>>(keys, E, hist);
    sel_hi_kernel<<<1, 1, 0, s>>>(hist, target, sel);
    (void)hipMemsetAsync(hist, 0, 65536 * sizeof(unsigned), s);
    hist_lo_kernel<<<nt, 256, 0, s>>>(keys, E, hist, sel);
    sel_lo_kernel<<<1, 1, 0, s>>>(hist, target, sel);
    eq_flags_kernel<<<nt, 256, 0, s>>>(keys, E, sel, flags);
    run_scan(flags, scanA, bsums, E, s);
    keep_flags_kernel<<<nt, 256, 0, s>>>(keys, E, sel, scanA, flags);
    run_scan(flags, scanB, bsums, E, s);
    compact_kernel<<<nt, 256, 0, s>>>(flags, scanB, E, keepIdx, remap);
    nb_pool_kernel<<<(target + 255) / 256, 256, 0, s>>>(keepIdx, remap, nb_old, E,
                                                        target, nb_new);
    int n = target * C;
    pool_x_kernel<<<(n + 255) / 256, 256, 0, s>>>(xin, keepIdx, target, C, xp);
}

extern "C" void kernel_launch(void* const* d_in, const int* in_sizes, int n_in,
                              void* d_out, int out_size, void* d_ws, size_t ws_size,
                              hipStream_t stream)
{
    const float* x  = (const float*)d_in[0];
    const int*   nb = (const int*)d_in[1];
    const float* W_e1 = (const float*)d_in[2],  *g_e1 = (const float*)d_in[3],  *b_e1 = (const float*)d_in[4];
    const float* W_e2 = (const float*)d_in[5],  *g_e2 = (const float*)d_in[6],  *b_e2 = (const float*)d_in[7];
    const float* W_e3 = (const float*)d_in[8],  *g_e3 = (const float*)d_in[9],  *b_e3 = (const float*)d_in[10];
    const float* W_e4 = (const float*)d_in[11], *g_e4 = (const float*)d_in[12], *b_e4 = (const float*)d_in[13];
    const float* W_d3 = (const float*)d_in[14], *g_d3 = (const float*)d_in[15], *b_d3 = (const float*)d_in[16];
    const float* W_d2 = (const float*)d_in[17], *g_d2 = (const float*)d_in[18], *b_d2 = (const float*)d_in[19];
    const float* W_d1 = (const float*)d_in[20], *g_d1 = (const float*)d_in[21], *b_d1 = (const float*)d_in[22];
    const float* W_head = (const float*)d_in[23];
    const float* bias_head = (const float*)d_in[24];

    const int E0 = in_sizes[0] / 5;   // 200000
    const int T1 = 1500, T2 = 750, T3 = 375;

    // bump allocator over d_ws
    char* base = (char*)d_ws;
    size_t used = 0;
    auto alloc = [&](size_t bytes) -> void* {
        bytes = (bytes + 255) & ~(size_t)255;
        void* r = base + used;
        used += bytes;
        return r;
    };

    float* e1    = (float*)alloc((size_t)E0 * 64 * 4);
    float* e1p   = (float*)alloc((size_t)T1 * 64 * 4);
    int*   nb1   = (int*)alloc((size_t)T1 * 4 * 4);
    int*   keep1 = (int*)alloc((size_t)T1 * 4);
    float* e2    = (float*)alloc((size_t)T1 * 128 * 4);
    float* e2p   = (float*)alloc((size_t)T2 * 128 * 4);
    int*   nb2   = (int*)alloc((size_t)T2 * 4 * 4);
    int*   keep2 = (int*)alloc((size_t)T2 * 4);
    float* e3    = (float*)alloc((size_t)T2 * 256 * 4);
    float* e3p   = (float*)alloc((size_t)T3 * 256 * 4);
    int*   nb3   = (int*)alloc((size_t)T3 * 4 * 4);
    int*   keep3 = (int*)alloc((size_t)T3 * 4);
    float* e4    = (float*)alloc((size_t)T3 * 512 * 4);
    float* d3cat = (float*)alloc((size_t)T2 * 768 * 4);
    float* d3o   = (float*)alloc((size_t)T2 * 256 * 4);
    float* d2cat = (float*)alloc((size_t)T1 * 384 * 4);
    float* d2o   = (float*)alloc((size_t)T1 * 128 * 4);
    float* d1cat = (float*)alloc((size_t)E0 * 192 * 4);
    float* d1o   = (float*)alloc((size_t)E0 * 64 * 4);
    float*    sums  = (float*)alloc(2 * 512 * 4);
    unsigned* keys  = (unsigned*)alloc((size_t)E0 * 4);
    unsigned* hist  = (unsigned*)alloc(65536 * 4);
    unsigned* sel   = (unsigned*)alloc(64);
    unsigned* flags = (unsigned*)alloc((size_t)E0 * 4);
    unsigned* scanA = (unsigned*)alloc((size_t)E0 * 4);
    unsigned* scanB = (unsigned*)alloc((size_t)E0 * 4);
    unsigned* bsums = (unsigned*)alloc(4096);
    int*      remap = (int*)alloc((size_t)E0 * 4);
    if (used > ws_size) return;   // workspace guard

    // ---- encoder ----
    run_conv<5, 64>(x, nb, E0, W_e1, g_e1, b_e1, e1, sums, stream);
    run_pool(e1, nb, E0, 64, T1, e1p, nb1, keep1, remap,
             keys, hist, sel, flags, scanA, scanB, bsums, stream);

    run_conv<64, 128>(e1p, nb1, T1, W_e2, g_e2, b_e2, e2, sums, stream);
    run_pool(e2, nb1, T1, 128, T2, e2p, nb2, keep2, remap,
             keys, hist, sel, flags, scanA, scanB, bsums, stream);

    run_conv<128, 256>(e2p, nb2, T2, W_e3, g_e3, b_e3, e3, sums, stream);
    run_pool(e3, nb2, T2, 256, T3, e3p, nb3, keep3, remap,
             keys, hist, sel, flags, scanA, scanB, bsums, stream);

    run_conv<256, 512>(e3p, nb3, T3, W_e4, g_e4, b_e4, e4, sums, stream);

    // ---- decoder ----
    {
        int n = T2 * 512;
        unpool_cat_kernel<<<(n + 255) / 256, 256, 0, stream>>>(e4, keep3, T3, T2, 512, 768, d3cat);
        int m = T2 * 256;
        copy_offset_kernel<<<(m + 255) / 256, 256, 0, stream>>>(e3, T2, 256, 768, 512, d3cat);
        run_conv<768, 256>(d3cat, nb2, T2, W_d3, g_d3, b_d3, d3o, sums, stream);
    }
    {
        int n = T1 * 256;
        unpool_cat_kernel<<<(n + 255) / 256, 256, 0, stream>>>(d3o, keep2, T2, T1, 256, 384, d2cat);
        int m = T1 * 128;
        copy_offset_kernel<<<(m + 255) / 256, 256, 0, stream>>>(e2, T1, 128, 384, 256, d2cat);
        run_conv<384, 128>(d2cat, nb1, T1, W_d2, g_d2, b_d2, d2o, sums, stream);
    }
    {
        int n = E0 * 128;
        unpool_cat_kernel<<<(n + 255) / 256, 256, 0, stream>>>(d2o, keep1, T1, E0, 128, 192, d1cat);
        int m = E0 * 64;
        copy_offset_kernel<<<(m + 255) / 256, 256, 0, stream>>>(e1, E0, 64, 192, 128, d1cat);
        run_conv<192, 64>(d1cat, nb, E0, W_d1, g_d1, b_d1, d1o, sums, stream);
    }

    head_kernel<<<(E0 + 255) / 256, 256, 0, stream>>>(d1o, W_head, bias_head, E0,
                                                      (float*)d_out);
    (void)in_sizes; (void)n_in; (void)out_size;
}